// MfWf_48816598286533
// MI455X (gfx1250) — compile-verified
//
#include <hip/hip_runtime.h>

// ---------------------------------------------------------------------------
// MfWf: per-(b,t,f) 5x5 complex solve + 5-tap complex filter.
// B=16, C=1, T=1000, F=481, Fd=96, N=5. coefs: (B, 30, T, 96, 2) f32.
// Memory-bound (~490 MB @ 23.3 TB/s => ~21us floor); fp32 scalar math,
// CDNA5 async global<->LDS path used for the bulk pass-through copy.
// ---------------------------------------------------------------------------

#define GAS __attribute__((address_space(1)))
#define LAS __attribute__((address_space(3)))

typedef int v4i __attribute__((vector_size(4 * sizeof(int))));

#if __has_builtin(__builtin_amdgcn_global_load_async_to_lds_b128) && \
    __has_builtin(__builtin_amdgcn_global_store_async_from_lds_b128)
#define USE_ASYNC_LDS_COPY 1
#else
#define USE_ASYNC_LDS_COPY 0
#endif

__device__ __forceinline__ void wait_async0() {
#if __has_builtin(__builtin_amdgcn_s_wait_asynccnt)
  __builtin_amdgcn_s_wait_asynccnt(0);
#else
  asm volatile("s_wait_asynccnt 0" ::: "memory");
#endif
}

struct cplx { float re, im; };

__device__ __forceinline__ cplx cmul(cplx a, cplx b) {
  cplx r;
  r.re = fmaf(a.re, b.re, -a.im * b.im);
  r.im = fmaf(a.re, b.im,  a.im * b.re);
  return r;
}
// acc - a*b
__device__ __forceinline__ cplx cnmsub(cplx acc, cplx a, cplx b) {
  acc.re = fmaf(-a.re, b.re, fmaf( a.im, b.im, acc.re));
  acc.im = fmaf(-a.re, b.im, fmaf(-a.im, b.re, acc.im));
  return acc;
}
__device__ __forceinline__ void cswap(bool s, cplx& a, cplx& b) {
  cplx t = a;
  a.re = s ? b.re : a.re;  a.im = s ? b.im : a.im;
  b.re = s ? t.re : b.re;  b.im = s ? t.im : b.im;
}

// ---------------------------------------------------------------------------
// Kernel 1: bulk copy spec -> out (whole buffer, 16B granules).
// Uses CDNA5 async global->LDS->global path (ASYNCcnt) when available:
// a VGPR-free streaming copy; 4 granules per lane in flight per iteration.
// ---------------------------------------------------------------------------
__global__ __launch_bounds__(256) void copy_async(const float4* __restrict__ src,
                                                  float4* __restrict__ dst,
                                                  long n4) {
  long tid    = (long)blockIdx.x * blockDim.x + threadIdx.x;
  long stride = (long)gridDim.x * blockDim.x;
#if USE_ASYNC_LDS_COPY
  __shared__ float4 lbuf[256 * 4];
  for (long base = tid; base < n4; base += stride * 4) {
#pragma unroll
    for (int k = 0; k < 4; ++k) {
      long i = base + (long)k * stride;
      if (i < n4) {
        __builtin_amdgcn_global_load_async_to_lds_b128(
            (GAS v4i*)(uintptr_t)(src + i),
            (LAS v4i*)(uintptr_t)&lbuf[threadIdx.x + k * 256], 0, 0);
      }
    }
    wait_async0();  // loads landed in LDS
#pragma unroll
    for (int k = 0; k < 4; ++k) {
      long i = base + (long)k * stride;
      if (i < n4) {
        __builtin_amdgcn_global_store_async_from_lds_b128(
            (GAS v4i*)(uintptr_t)(dst + i),
            (LAS v4i*)(uintptr_t)&lbuf[threadIdx.x + k * 256], 0, 0);
      }
    }
    wait_async0();  // stores drained before LDS slots are reused
  }
#else
  for (long i = tid; i < n4; i += stride) dst[i] = src[i];
#endif
}

// ---------------------------------------------------------------------------
// Kernel 2: one thread per (b,t,f<96) site.
//   A = Rxx (5x5 complex), rhs = rss; solve A w = rhs via in-register
//   Gaussian elimination with branchless partial pivoting; then
//   y = sum_n x[b,t-4+n,f] * w[n]; overwrite out at f<96.
// ---------------------------------------------------------------------------
__global__ __launch_bounds__(256) void mfwf_solve(const float* __restrict__ spec,
                                                  const float* __restrict__ coefs,
                                                  float* __restrict__ out) {
  constexpr int FD = 96, T = 1000, F = 481;
  int g = blockIdx.x * 256 + threadIdx.x;
  constexpr int total = 16 * T * FD;
  if (g >= total) return;
  int f = g % FD;
  int t = (g / FD) % T;
  int b = g / (FD * T);

  // coefs element (b,j,t,f) as float2: ((b*30 + j)*T + t)*FD + f
  const float2* cf = reinterpret_cast<const float2*>(coefs);
  const size_t jstride = (size_t)T * FD;
  size_t cbase = (((size_t)b * 30) * T + (size_t)t) * FD + (size_t)f;

  cplx A[5][5], rhs[5], pinv[5], w[5];
#pragma unroll
  for (int j = 0; j < 25; ++j) {
    float2 v = cf[cbase + (size_t)j * jstride];
    A[j / 5][j % 5] = {v.x, v.y};            // Rxx[n][m] = c[b, n*5+m, t, f]
  }
#pragma unroll
  for (int n = 0; n < 5; ++n) {
    float2 v = cf[cbase + (size_t)(25 + n) * jstride];
    rhs[n] = {v.x, v.y};                      // rss[n] = c[b, 25+n, t, f]
  }

  // Gaussian elimination with partial pivoting (all static indices -> VGPRs).
#pragma unroll
  for (int k = 0; k < 5; ++k) {
#pragma unroll
    for (int r = k + 1; r < 5; ++r) {        // bubble largest |A[r][k]| to row k
      float mk = fmaf(A[k][k].re, A[k][k].re, A[k][k].im * A[k][k].im);
      float mr = fmaf(A[r][k].re, A[r][k].re, A[r][k].im * A[r][k].im);
      bool sw = mr > mk;
#pragma unroll
      for (int j = k; j < 5; ++j) cswap(sw, A[k][j], A[r][j]);
      cswap(sw, rhs[k], rhs[r]);
    }
    float d  = fmaf(A[k][k].re, A[k][k].re, A[k][k].im * A[k][k].im);
    float rd = 1.0f / d;
    pinv[k] = {A[k][k].re * rd, -A[k][k].im * rd};   // 1 / pivot
#pragma unroll
    for (int r = k + 1; r < 5; ++r) {
      cplx m = cmul(A[r][k], pinv[k]);
#pragma unroll
      for (int j = k + 1; j < 5; ++j) A[r][j] = cnmsub(A[r][j], m, A[k][j]);
      rhs[r] = cnmsub(rhs[r], m, rhs[k]);
    }
  }
  // Back substitution.
#pragma unroll
  for (int k = 4; k >= 0; --k) {
    cplx s = rhs[k];
#pragma unroll
    for (int j = k + 1; j < 5; ++j) s = cnmsub(s, A[k][j], w[j]);
    w[k] = cmul(s, pinv[k]);
  }

  // 5-tap complex filter over zero-padded time axis.
  const float2* sp = reinterpret_cast<const float2*>(spec);
  cplx y = {0.f, 0.f};
#pragma unroll
  for (int n = 0; n < 5; ++n) {
    int tt = t - 4 + n;
    if (tt >= 0) {
      float2 v = sp[((size_t)b * T + (size_t)tt) * F + (size_t)f];
      cplx x = {v.x, v.y};
      y.re = fmaf(x.re, w[n].re, fmaf(-x.im, w[n].im, y.re));
      y.im = fmaf(x.re, w[n].im, fmaf( x.im, w[n].re, y.im));
    }
  }

  float2* op = reinterpret_cast<float2*>(out);
  op[((size_t)b * T + (size_t)t) * F + (size_t)f] = make_float2(y.re, y.im);
}

// ---------------------------------------------------------------------------
extern "C" void kernel_launch(void* const* d_in, const int* in_sizes, int n_in,
                              void* d_out, int out_size, void* d_ws, size_t ws_size,
                              hipStream_t stream) {
  const float* spec  = (const float*)d_in[0];   // (16,1,1000,481,2) f32
  const float* coefs = (const float*)d_in[1];   // (16,30,1000,96,2) f32
  float* out = (float*)d_out;                   // same shape as spec
  (void)in_sizes; (void)n_in; (void)d_ws; (void)ws_size;

  // 1) stream the whole spectrogram through (f>=96 region is final here).
  long n4 = (long)out_size / 4;                 // 3,848,000 float4 granules
  copy_async<<<1024, 256, 0, stream>>>((const float4*)spec, (float4*)out, n4);

  // 2) solve + filter, overwriting f<96 (stream order guarantees after copy).
  constexpr int total = 16 * 1000 * 96;
  mfwf_solve<<<(total + 255) / 256, 256, 0, stream>>>(spec, coefs, out);
}